// GateFrameSelector_39505109188839
// MI455X (gfx1250) — compile-verified
//
#include <hip/hip_runtime.h>
#include <hip/hip_bf16.h>
#include <math.h>

// ---------------------------------------------------------------------------
// GateFrameSelector for MI455X (gfx1250, wave32).
//
// Roofline: mean over tokens streams 1.03 GB (>= ~44us @ 23.3 TB/s) and
// dominates. GEMMs are ~4 GFLOP total -> use full-precision f32 WMMA
// (V_WMMA_F32_16X16X4_F32), no reason to quantize. Pool kernel uses
// non-temporal 128b loads so the 1 GB stream doesn't pollute L2.
// ---------------------------------------------------------------------------

#define N_FRAMES 512
#define TOKENS   196
#define EMBED    2560
#define HIDDEN   512
#define LN_EPS   1e-5f
#define F2F_THRD 0.98f
#define F2T_THRD (-1.0f)
#define MAX_SEL  32

typedef __attribute__((ext_vector_type(2))) float v2f;
typedef __attribute__((ext_vector_type(4))) float v4f;
typedef __attribute__((ext_vector_type(8))) float v8f;

// ---------------------------------------------------------------------------
// 1) pooled[n][e] = mean_t image[n][t][e].  One block per frame, 640 threads,
//    each thread owns one float4 column group; NT loads (streaming 1 GB).
// ---------------------------------------------------------------------------
__global__ void pool_mean_kernel(const float* __restrict__ img,
                                 float* __restrict__ pooled) {
  const int n  = blockIdx.x;
  const int e4 = threadIdx.x;                       // 0..639
  const float* base = img + (size_t)n * TOKENS * EMBED + (size_t)e4 * 4;
  float ax = 0.f, ay = 0.f, az = 0.f, aw = 0.f;
#pragma unroll 4
  for (int t = 0; t < TOKENS; ++t) {
    v4f v = __builtin_nontemporal_load((const v4f*)(base + (size_t)t * EMBED));
    ax += v[0]; ay += v[1]; az += v[2]; aw += v[3];
  }
  const float s = 1.0f / (float)TOKENS;
  float* o = pooled + (size_t)n * EMBED + (size_t)e4 * 4;
  o[0] = ax * s; o[1] = ay * s; o[2] = az * s; o[3] = aw * s;
}

// ---------------------------------------------------------------------------
// 2) LayerNorm rows. Blocks 0..511: pooled rows (in-place) + L2-normalized
//    copy into pn. Block 512: text row -> txt_ln.
// ---------------------------------------------------------------------------
__global__ void ln_kernel(float* __restrict__ pooled,
                          const float* __restrict__ txt_in,
                          const float* __restrict__ lt_g, const float* __restrict__ lt_b,
                          const float* __restrict__ ll_g, const float* __restrict__ ll_b,
                          float* __restrict__ pn, float* __restrict__ txt_ln) {
  __shared__ float red[256];
  __shared__ float s_mean, s_rstd, s_rn;
  const int row = blockIdx.x;
  const int tid = threadIdx.x;
  const bool is_txt = (row == N_FRAMES);
  const float* x = is_txt ? txt_in : pooled + (size_t)row * EMBED;
  const float* g = is_txt ? lt_g : ll_g;
  const float* b = is_txt ? lt_b : ll_b;
  float* y_out   = is_txt ? txt_ln : pooled + (size_t)row * EMBED;

  float lx[10];
  float s = 0.f, ss = 0.f;
#pragma unroll
  for (int i = 0; i < 10; ++i) {
    float v = x[tid + i * 256];
    lx[i] = v; s += v; ss += v * v;
  }
  red[tid] = s; __syncthreads();
  for (int st = 128; st > 0; st >>= 1) { if (tid < st) red[tid] += red[tid + st]; __syncthreads(); }
  const float sum = red[0];
  __syncthreads();
  red[tid] = ss; __syncthreads();
  for (int st = 128; st > 0; st >>= 1) { if (tid < st) red[tid] += red[tid + st]; __syncthreads(); }
  if (tid == 0) {
    float mean = sum * (1.0f / EMBED);
    float var  = red[0] * (1.0f / EMBED) - mean * mean;
    s_mean = mean;
    s_rstd = rsqrtf(var + LN_EPS);
  }
  __syncthreads();
  const float mean = s_mean, rstd = s_rstd;

  float ly[10];
  float nrm = 0.f;
#pragma unroll
  for (int i = 0; i < 10; ++i) {
    int e = tid + i * 256;
    float v = (lx[i] - mean) * rstd * g[e] + b[e];
    ly[i] = v;
    y_out[e] = v;
    nrm += v * v;
  }
  if (!is_txt) {
    __syncthreads();
    red[tid] = nrm; __syncthreads();
    for (int st = 128; st > 0; st >>= 1) { if (tid < st) red[tid] += red[tid + st]; __syncthreads(); }
    if (tid == 0) s_rn = 1.0f / fmaxf(sqrtf(red[0]), 1e-8f);
    __syncthreads();
    const float rn = s_rn;
#pragma unroll
    for (int i = 0; i < 10; ++i)
      pn[(size_t)row * EMBED + tid + i * 256] = ly[i] * rn;
  }
}

// ---------------------------------------------------------------------------
// 3) h1 = relu(fused @ W1 + b1).  fused[m][k] = k<E ? txt_ln[k]
//    : pooled_ln[m][k-E].  One wave per 16x16 C tile; f32 WMMA, K-step 4.
//    1024 waves total (32x32 tiles): EXEC is all-ones everywhere.
// ---------------------------------------------------------------------------
__global__ void gemm1_wmma_kernel(const float* __restrict__ txt_ln,
                                  const float* __restrict__ pooled_ln,
                                  const float* __restrict__ W1,
                                  const float* __restrict__ b1,
                                  float* __restrict__ h1) {
  const int gid  = blockIdx.x * blockDim.x + threadIdx.x;
  const int wave = gid >> 5;
  const int lane = gid & 31;
  const int m0   = (wave >> 5) << 4;   // 0..496
  const int n0   = (wave & 31) << 4;   // 0..496
  const int half = lane >> 4;          // K sub-pair select (0: K=0,1  1: K=2,3)
  const int l    = lane & 15;
  const int row  = m0 + l;             // A-matrix row for this lane
  const int col  = n0 + l;             // B/C column for this lane

  v8f acc = {};
  for (int k = 0; k < 2 * EMBED; k += 4) {
    const int ka = k + 2 * half;
    v2f a, bfr;
    if (ka < EMBED) {                  // uniform across wave (boundary % 4 == 0)
      a[0] = txt_ln[ka];
      a[1] = txt_ln[ka + 1];
    } else {
      const float* p = pooled_ln + (size_t)row * EMBED + (ka - EMBED);
      a[0] = p[0];
      a[1] = p[1];
    }
    bfr[0] = W1[(size_t)ka * HIDDEN + col];
    bfr[1] = W1[(size_t)(ka + 1) * HIDDEN + col];
    acc = __builtin_amdgcn_wmma_f32_16x16x4_f32(false, a, false, bfr,
                                                (short)0, acc, false, false);
  }
  const float bias = b1[col];
#pragma unroll
  for (int r = 0; r < 8; ++r) {
    const int orow = m0 + r + 8 * half;
    h1[(size_t)orow * HIDDEN + col] = fmaxf(acc[r] + bias, 0.0f);
  }
}

// ---------------------------------------------------------------------------
// 4) gates[m] = sigmoid(h1[m] . W2 + b2).  One block, thread per frame.
// ---------------------------------------------------------------------------
__global__ void gates_kernel(const float* __restrict__ h1,
                             const float* __restrict__ W2,
                             const float* __restrict__ b2,
                             float* __restrict__ gates_ws,
                             float* __restrict__ out_gates) {
  const int m = threadIdx.x;
  const float* r = h1 + (size_t)m * HIDDEN;
  float s = 0.f;
#pragma unroll 8
  for (int j = 0; j < HIDDEN; ++j) s += r[j] * W2[j];
  const float gv = 1.0f / (1.0f + __expf(-(s + b2[0])));
  gates_ws[m]  = gv;
  out_gates[m] = gv;
}

// ---------------------------------------------------------------------------
// 5) f2f = pn @ pn^T (cosine sim).  Same f32 WMMA tiling, K = 2560.
//    B fragment = pn row of the *column* index (transpose-free).
// ---------------------------------------------------------------------------
__global__ void f2f_wmma_kernel(const float* __restrict__ pn,
                                float* __restrict__ f2f) {
  const int gid  = blockIdx.x * blockDim.x + threadIdx.x;
  const int wave = gid >> 5;
  const int lane = gid & 31;
  const int m0   = (wave >> 5) << 4;
  const int n0   = (wave & 31) << 4;
  const int half = lane >> 4;
  const int l    = lane & 15;
  const float* pa = pn + (size_t)(m0 + l) * EMBED;
  const float* pb = pn + (size_t)(n0 + l) * EMBED;

  v8f acc = {};
  for (int k = 0; k < EMBED; k += 4) {
    const int ka = k + 2 * half;
    v2f a, bfr;
    a[0]   = pa[ka];     a[1]   = pa[ka + 1];
    bfr[0] = pb[ka];     bfr[1] = pb[ka + 1];
    acc = __builtin_amdgcn_wmma_f32_16x16x4_f32(false, a, false, bfr,
                                                (short)0, acc, false, false);
  }
#pragma unroll
  for (int r = 0; r < 8; ++r) {
    const int orow = m0 + r + 8 * half;
    f2f[(size_t)orow * N_FRAMES + n0 + l] = acc[r];
  }
}

// ---------------------------------------------------------------------------
// 6) Greedy selection: bitonic argsort (desc) of gates, then serial scan with
//    parallel visited-update via the f2f row of each taken frame.
// ---------------------------------------------------------------------------
__global__ void select_kernel(const float* __restrict__ gates,
                              const float* __restrict__ f2f,
                              float* __restrict__ out_sel) {
  __shared__ float skey[N_FRAMES];
  __shared__ int   sidx[N_FRAMES];
  __shared__ unsigned char visited[N_FRAMES];
  __shared__ int s_cur, s_take, s_count;

  const int tid = threadIdx.x;
  skey[tid] = gates[tid];
  sidx[tid] = tid;
  visited[tid] = 0;
  if (tid == 0) s_count = 0;
  __syncthreads();

  // Bitonic sort, descending by key.
  for (int k = 2; k <= N_FRAMES; k <<= 1) {
    for (int j = k >> 1; j > 0; j >>= 1) {
      const int ixj = tid ^ j;
      if (ixj > tid) {
        const bool up = ((tid & k) == 0);
        if ((skey[tid] < skey[ixj]) == up) {
          float tk = skey[tid]; skey[tid] = skey[ixj]; skey[ixj] = tk;
          int   ti = sidx[tid]; sidx[tid] = sidx[ixj]; sidx[ixj] = ti;
        }
      }
      __syncthreads();
    }
  }

  int sel = 0;
  for (int i = 0; i < N_FRAMES; ++i) {
    if (tid == 0) {
      const int   cur = sidx[i];
      const float sc  = skey[i];
      int take = (s_count < MAX_SEL) && (sc >= F2T_THRD) && (visited[cur] == 0);
      s_cur = cur; s_take = take;
      if (take) s_count++;
    }
    __syncthreads();
    if (s_take) {
      if (f2f[(size_t)s_cur * N_FRAMES + tid] > F2F_THRD) visited[tid] = 1;
      if (tid == s_cur) sel = 1;
    }
    __syncthreads();
  }
  out_sel[tid] = (float)sel;
}

// ---------------------------------------------------------------------------
extern "C" void kernel_launch(void* const* d_in, const int* in_sizes, int n_in,
                              void* d_out, int out_size, void* d_ws, size_t ws_size,
                              hipStream_t stream) {
  const float* img  = (const float*)d_in[0];
  const float* txt  = (const float*)d_in[1];
  const float* lt_g = (const float*)d_in[2];
  const float* lt_b = (const float*)d_in[3];
  const float* ll_g = (const float*)d_in[4];
  const float* ll_b = (const float*)d_in[5];
  const float* W1   = (const float*)d_in[6];
  const float* b1   = (const float*)d_in[7];
  const float* W2   = (const float*)d_in[8];
  const float* b2   = (const float*)d_in[9];

  float* ws      = (float*)d_ws;
  float* pooled  = ws;                                   // 512*2560 (raw -> LN in-place)
  float* pn      = pooled + (size_t)N_FRAMES * EMBED;    // 512*2560
  float* txt_ln  = pn + (size_t)N_FRAMES * EMBED;        // 2560
  float* h1      = txt_ln + EMBED;                       // 512*512
  float* f2f     = h1 + (size_t)N_FRAMES * HIDDEN;       // 512*512
  float* gates   = f2f + (size_t)N_FRAMES * N_FRAMES;    // 512
  float* out     = (float*)d_out;                        // [selected(512), gates(512)]

  pool_mean_kernel<<<N_FRAMES, 640, 0, stream>>>(img, pooled);
  ln_kernel<<<N_FRAMES + 1, 256, 0, stream>>>(pooled, txt, lt_g, lt_b, ll_g, ll_b, pn, txt_ln);
  gemm1_wmma_kernel<<<128, 256, 0, stream>>>(txt_ln, pooled, W1, b1, h1);
  gates_kernel<<<1, N_FRAMES, 0, stream>>>(h1, W2, b2, gates, out + N_FRAMES);
  f2f_wmma_kernel<<<128, 256, 0, stream>>>(pn, f2f);
  select_kernel<<<1, N_FRAMES, 0, stream>>>(gates, f2f, out);
}